// PretrainNeck_53755810677394
// MI455X (gfx1250) — compile-verified
//
#include <hip/hip_runtime.h>
#include <hip/hip_bf16.h>

typedef float v2f __attribute__((ext_vector_type(2)));
typedef float v4f __attribute__((ext_vector_type(4)));
typedef float v8f __attribute__((ext_vector_type(8)));

#define N_SAMP 32
#define M_DIM  2
#define C_DIM  256
#define SEG    6400          // T*V = 256*25 contiguous floats per (n,m,c)
#define CHUNKS 50            // SEG / 128
#define WAVES_PER_BLOCK 8
#define BLOCK (WAVES_PER_BLOCK * 32)

// out[n,c] = (1/20) * sum_{m,t,v} x[n,m,c,t,v]
// One wave32 per (n,c). Streaming fp32 reduction routed through
// v_wmma_f32_16x16x4_f32 with an all-ones A matrix (row-sum-by-matmul).
__global__ __launch_bounds__(BLOCK)
void neck_reduce_wmma(const float* __restrict__ x, float* __restrict__ out) {
    const int wave = blockIdx.x * WAVES_PER_BLOCK + (threadIdx.x >> 5);
    const int lane = threadIdx.x & 31;
    const int n = wave >> 8;          // [0,32)
    const int c = wave & 255;         // [0,256)

    // segment bases for m=0 and m=1 (each 6400 contiguous floats)
    const float* p0 = x + (size_t)((n * M_DIM + 0) * C_DIM + c) * SEG + 4 * lane;
    const float* p1 = x + (size_t)((n * M_DIM + 1) * C_DIM + c) * SEG + 4 * lane;

    v2f a;                            // A = ones(16x4): 2 VGPRs, all lanes = 1.0
    a.x = 1.0f; a.y = 1.0f;
    v8f acc = {};                     // D/C accumulator (16x16 f32)

#pragma unroll 2
    for (int i = 0; i < CHUNKS; ++i) {
        // 128 contiguous floats per wave per stream: lane L -> floats [4L..4L+3]
        v4f f0 = __builtin_nontemporal_load((const v4f*)(p0 + i * 128));
        v4f f1 = __builtin_nontemporal_load((const v4f*)(p1 + i * 128));

        v2f b;
        b.x = f0.x; b.y = f0.y;       // B tile: 2 VGPRs = 64 lane-floats
        acc = __builtin_amdgcn_wmma_f32_16x16x4_f32(false, a, false, b,
                                                    (short)0, acc, false, false);
        b.x = f0.z; b.y = f0.w;
        acc = __builtin_amdgcn_wmma_f32_16x16x4_f32(false, a, false, b,
                                                    (short)0, acc, false, false);
        b.x = f1.x; b.y = f1.y;
        acc = __builtin_amdgcn_wmma_f32_16x16x4_f32(false, a, false, b,
                                                    (short)0, acc, false, false);
        b.x = f1.z; b.y = f1.w;
        acc = __builtin_amdgcn_wmma_f32_16x16x4_f32(false, a, false, b,
                                                    (short)0, acc, false, false);
    }

    // A == ones => every row of D identical. acc[0] holds D[0, lane%16]
    // (duplicated across both lane halves). Sum over 32 lanes = 2 * sum_j D[0,j].
    float s = acc[0];
    for (int m = 16; m > 0; m >>= 1)
        s += __shfl_xor(s, m, 32);

    if (lane == 0)
        out[n * C_DIM + c] = s * (1.0f / 40.0f);  // /2 dup, /2 mean_m, /10 mean_Kl
}

extern "C" void kernel_launch(void* const* d_in, const int* in_sizes, int n_in,
                              void* d_out, int out_size, void* d_ws, size_t ws_size,
                              hipStream_t stream) {
    const float* x = (const float*)d_in[0];   // [32,2,256,256,25] fp32
    // d_in[1..3] (protos) provably do not affect the output (full-partition
    // segment sums make the VQ assignments irrelevant) -> unused.
    float* out = (float*)d_out;               // [32,256] fp32

    const int total_waves = N_SAMP * C_DIM;   // 8192
    const int grid = total_waves / WAVES_PER_BLOCK;  // 1024 blocks
    neck_reduce_wmma<<<grid, BLOCK, 0, stream>>>(x, out);
}